// GraphIsomorphismNetwork_34574486732951
// MI455X (gfx1250) — compile-verified
//
#include <hip/hip_runtime.h>

#define N_NODES 50000
#define N_EDGES 600000
#define HID     128
#define HH      (HID * HID)
#define N_GRAPHS 64
#define PITCH   132   // 128 + 4 pad: LDS column reads hit distinct banks

typedef __attribute__((ext_vector_type(2))) float v2f;
typedef __attribute__((ext_vector_type(8))) float v8f;

__device__ __forceinline__ float sp(float v) {
    // numerically stable softplus
    if (v > 20.0f) return v;
    return log1pf(__expf(v));
}

// ---------------------------------------------------------------- copy agg = x
__global__ __launch_bounds__(256) void copy_kernel(const float* __restrict__ src,
                                                   float* __restrict__ dst, int n4) {
    int i = blockIdx.x * 256 + threadIdx.x;
    if (i < n4) ((float4*)dst)[i] = ((const float4*)src)[i];
}

// -------------------------------------------- edge scatter: agg += softplus(x[src]+ea)
__global__ __launch_bounds__(256) void scatter_kernel(const float* __restrict__ x,
                                                      const float* __restrict__ ea,
                                                      const int* __restrict__ srcI,
                                                      const int* __restrict__ dstI,
                                                      float* __restrict__ agg, int E) {
    int gid = blockIdx.x * 256 + threadIdx.x;
    int e = gid >> 5;                 // one edge per 32 lanes (wave32)
    if (e >= E) return;
    int c = (gid & 31) << 2;          // 4 floats per lane -> 128 feats per wave
    int s = srcI[e];
    int d = dstI[e];
    const float4 xv = *(const float4*)(x  + (size_t)s * HID + c);
    const float4 ev = *(const float4*)(ea + (size_t)e * HID + c);
    float m0 = sp(xv.x + ev.x);
    float m1 = sp(xv.y + ev.y);
    float m2 = sp(xv.z + ev.z);
    float m3 = sp(xv.w + ev.w);
    float* a = agg + (size_t)d * HID + c;
    atomicAdd(a + 0, m0);
    atomicAdd(a + 1, m1);
    atomicAdd(a + 2, m2);
    atomicAdd(a + 3, m3);
}

// ------------------------------------- fused MLP: out = (softplus?)(softplus(agg@W1^T+b1)@W2^T+b2)
// 256 threads = 8 waves; block owns a 16-row node tile; wave w owns output cols [16w,16w+16).
__global__ __launch_bounds__(256) void mlp_kernel(const float* __restrict__ agg,
                                                  const float* __restrict__ W1,
                                                  const float* __restrict__ B1,
                                                  const float* __restrict__ W2,
                                                  const float* __restrict__ B2,
                                                  float* __restrict__ out,
                                                  int nTiles, int interSoftplus) {
    __shared__ float xs[16 * PITCH];
    __shared__ float hs[16 * PITCH];
    const int tid  = threadIdx.x;
    const int wave = tid >> 5;
    const int lane = tid & 31;
    const int half = lane >> 4;     // 0: lanes 0-15, 1: lanes 16-31
    const int l16  = lane & 15;
    const int n0   = wave * 16;     // output-column tile base

    for (int t = blockIdx.x; t < nTiles; t += gridDim.x) {
        const int row0 = t * 16;

        // stage 16x128 x-tile (coalesced)
        #pragma unroll
        for (int i = 0; i < 8; ++i) {
            int idx = tid + i * 256;
            int r = idx >> 7, c = idx & 127;
            xs[r * PITCH + c] = agg[(size_t)(row0 + r) * HID + c];
        }
        __syncthreads();

        // ---- GEMM1: h = softplus(x @ W1^T + b1) ----
        // A (16x4 f32): VGPR0 = {K0 | K2}, VGPR1 = {K1 | K3}  (ISA 7.12.2)
        // B (4x16 f32): mirrored; lane l16 supplies column n0+l16 of the tile
        v8f acc = {0.f, 0.f, 0.f, 0.f, 0.f, 0.f, 0.f, 0.f};
        const float* w1r = W1 + (size_t)(n0 + l16) * HID;   // row n of W1 == col n of B
        for (int k0 = 0; k0 < HID; k0 += 4) {
            v2f a, b;
            a.x = xs[l16 * PITCH + k0 + 2 * half];
            a.y = xs[l16 * PITCH + k0 + 2 * half + 1];
            b.x = w1r[k0 + 2 * half];
            b.y = w1r[k0 + 2 * half + 1];
            acc = __builtin_amdgcn_wmma_f32_16x16x4_f32(false, a, false, b,
                                                        (short)0, acc, false, false);
        }
        {
            float bias = B1[n0 + l16];
            #pragma unroll
            for (int v = 0; v < 8; ++v) {
                int r = v + half * 8;               // C/D: VGPR v -> row v (lanes<16) / v+8
                hs[r * PITCH + n0 + l16] = sp(acc[v] + bias);
            }
        }
        __syncthreads();

        // ---- GEMM2: y = h @ W2^T + b2 ----
        v8f acc2 = {0.f, 0.f, 0.f, 0.f, 0.f, 0.f, 0.f, 0.f};
        const float* w2r = W2 + (size_t)(n0 + l16) * HID;
        for (int k0 = 0; k0 < HID; k0 += 4) {
            v2f a, b;
            a.x = hs[l16 * PITCH + k0 + 2 * half];
            a.y = hs[l16 * PITCH + k0 + 2 * half + 1];
            b.x = w2r[k0 + 2 * half];
            b.y = w2r[k0 + 2 * half + 1];
            acc2 = __builtin_amdgcn_wmma_f32_16x16x4_f32(false, a, false, b,
                                                         (short)0, acc2, false, false);
        }
        {
            float bias = B2[n0 + l16];
            #pragma unroll
            for (int v = 0; v < 8; ++v) {
                int r = v + half * 8;
                float val = acc2[v] + bias;
                if (interSoftplus) val = sp(val);
                out[(size_t)(row0 + r) * HID + n0 + l16] = val;
            }
        }
        __syncthreads();   // xs/hs reused next tile
    }
}

// ---------------------------------------------------------------- pooling
__global__ __launch_bounds__(256) void zero_kernel(float* __restrict__ p, int n) {
    int i = blockIdx.x * 256 + threadIdx.x;
    if (i < n) p[i] = 0.0f;
}

__global__ __launch_bounds__(256) void pool_kernel(const float* __restrict__ nf,
                                                   const int* __restrict__ batch,
                                                   float* __restrict__ gf, int total) {
    int i = blockIdx.x * 256 + threadIdx.x;
    if (i >= total) return;
    int n = i >> 7;
    int f = i & 127;
    atomicAdd(gf + (size_t)batch[n] * HID + f, nf[i]);
}

// ---------------------------------------------------------------- launch
extern "C" void kernel_launch(void* const* d_in, const int* in_sizes, int n_in,
                              void* d_out, int out_size, void* d_ws, size_t ws_size,
                              hipStream_t stream) {
    const float* node_attr = (const float*)d_in[0];
    const float* edge_attr = (const float*)d_in[1];
    const int*   edge_idx  = (const int*)d_in[2];
    const int*   batch     = (const int*)d_in[3];
    const float* W1        = (const float*)d_in[4];
    const float* B1        = (const float*)d_in[5];
    const float* W2        = (const float*)d_in[6];
    const float* B2        = (const float*)d_in[7];
    (void)in_sizes; (void)n_in; (void)out_size; (void)ws_size;

    const int* srcI = edge_idx;            // edge_index[0]
    const int* dstI = edge_idx + N_EDGES;  // edge_index[1]

    float* gf = (float*)d_out;                       // [64,128]
    float* nf = gf + N_GRAPHS * HID;                 // [50000,128] node_feature output
    float* xbuf = (float*)d_ws;                      // intermediate x
    float* agg  = xbuf + (size_t)N_NODES * HID;      // agg / MLP input

    const int n4      = N_NODES * HID / 4;           // 1.6M float4
    const int cpGrid  = (n4 + 255) / 256;
    const int scGrid  = (N_EDGES * 32 + 255) / 256;  // 75000
    const int nTiles  = N_NODES / 16;                // 3125 (exact)
    const int plGrid  = (N_NODES * HID + 255) / 256;

    for (int l = 0; l < 3; ++l) {
        const float* xin = (l == 0) ? node_attr : xbuf;
        float* xout = (l == 2) ? nf : xbuf;

        copy_kernel<<<cpGrid, 256, 0, stream>>>(xin, agg, n4);
        scatter_kernel<<<scGrid, 256, 0, stream>>>(xin, edge_attr, srcI, dstI, agg, N_EDGES);
        mlp_kernel<<<nTiles, 256, 0, stream>>>(agg,
                                               W1 + (size_t)l * HH, B1 + (size_t)l * HID,
                                               W2 + (size_t)l * HH, B2 + (size_t)l * HID,
                                               xout, nTiles, (l < 2) ? 1 : 0);
    }

    zero_kernel<<<(N_GRAPHS * HID + 255) / 256, 256, 0, stream>>>(gf, N_GRAPHS * HID);
    pool_kernel<<<plGrid, 256, 0, stream>>>(nf, batch, gf, N_NODES * HID);
}